// DynamicFilter_61040075210841
// MI455X (gfx1250) — compile-verified
//
#include <hip/hip_runtime.h>

// ---------------------------------------------------------------------------
// DynamicFilter for MI455X (gfx1250, wave32, WMMA).
// Precision: full fp32 via V_WMMA_F32_16X16X4_F32 (workload is bandwidth/L2
// bound, ~103 GFLOP vs ~200MB touched -> keep reference fp32 accuracy).
// Pipeline:
//   1) q     = x @ Wq                        (WMMA f32 GEMM, 32x32/wave)
//   2) xavg  = cumsum(q)/t                   (per-column serial scan)
//   3) g     = relu(xavg @ Wg + bg)          (WMMA f32 GEMM + bias/relu)
//   4) S[i]  = sum_{k=L-1-i}^{L-1} g[i,k]    (wave suffix-sum) + 1e-8
//   5) tmp   = (shifted-triangular attn) @ q (WMMA, gathered A, 16x32/wave,
//              unmasked bulk loop + masked diagonal tail -> EXEC stays full
//              in the hot loop, no per-load branches)
//   6) out   = tmp @ Wc                      (WMMA f32 GEMM)
// Workspace: q(32MB) + xavg/tmp(32MB, reused) + g(64MB) + S(32KB) ~= 128MB.
// ---------------------------------------------------------------------------

typedef __attribute__((ext_vector_type(2))) float v2f;
typedef __attribute__((ext_vector_type(8))) float v8f;

#define D_DIM 1024
#define LSEQ  2048

// ---------------------------------------------------------------------------
// fp32 WMMA GEMM:  C[M,N] = act(A[M,K] @ B[K,N] (+ bias))
// One wave -> 32x32 of C (2x2 grid of 16x16 WMMA tiles); 4 waves (2x2) per
// block -> 64x64 block tile. A fragments loaded as 8B v2f (K-pairs are
// contiguous and even-aligned). Each A frag feeds 2 WMMAs, each B frag 2.
// Fragment layouts (f32 16x16x4):
//   A (16x4): lanes 0-15 rows M0..15 K0/K1; lanes 16-31 same rows K2/K3.
//   B (4x16): lanes 0-15 cols N0..15 K0/K1; lanes 16-31 same cols K2/K3.
//   C (16x16): VGPR r = row r (lanes 0-15) / row r+8 (lanes 16-31).
// ---------------------------------------------------------------------------
template <bool RELU_BIAS>
__global__ __launch_bounds__(128) void gemm_f32_wmma(
    const float* __restrict__ A, const float* __restrict__ Bm,
    const float* __restrict__ bias, float* __restrict__ C,
    int M, int N, int K)
{
    const int lane = threadIdx.x & 31;
    const int wave = threadIdx.x >> 5;
    const int half = lane >> 4;     // 0: K0/K1 half, 1: K2/K3 half
    const int l16  = lane & 15;
    const int koff = half * 2;

    const int m0 = blockIdx.y * 64 + (wave >> 1) * 32;
    const int n0 = blockIdx.x * 64 + (wave & 1) * 32;

    const float* Arow0 = A + (size_t)(m0 + l16) * K + koff;
    const float* Arow1 = A + (size_t)(m0 + 16 + l16) * K + koff;
    const float* Bcol0 = Bm + n0 + l16;
    const float* Bcol1 = Bm + n0 + 16 + l16;

    v8f acc00 = {}, acc01 = {}, acc10 = {}, acc11 = {};
    for (int k0 = 0; k0 < K; k0 += 4) {
        v2f a0 = *(const v2f*)(Arow0 + k0);       // 8B load, even-aligned
        v2f a1 = *(const v2f*)(Arow1 + k0);
        v2f b0, b1;
        size_t r0 = (size_t)(k0 + koff) * N;
        size_t r1 = r0 + N;
        b0.x = Bcol0[r0];  b0.y = Bcol0[r1];
        b1.x = Bcol1[r0];  b1.y = Bcol1[r1];
        acc00 = __builtin_amdgcn_wmma_f32_16x16x4_f32(false, a0, false, b0, (short)0, acc00, false, false);
        acc01 = __builtin_amdgcn_wmma_f32_16x16x4_f32(false, a0, false, b1, (short)0, acc01, false, false);
        acc10 = __builtin_amdgcn_wmma_f32_16x16x4_f32(false, a1, false, b0, (short)0, acc10, false, false);
        acc11 = __builtin_amdgcn_wmma_f32_16x16x4_f32(false, a1, false, b1, (short)0, acc11, false, false);
    }

    float bv0 = 0.0f, bv1 = 0.0f;
    if (RELU_BIAS) {
        bv0 = bias[n0 + l16];
        bv1 = bias[n0 + 16 + l16];
    }
#pragma unroll
    for (int r = 0; r < 8; ++r) {
        int row0 = m0 + r + half * 8;
        int row1 = row0 + 16;
        float v00 = acc00[r], v01 = acc01[r], v10 = acc10[r], v11 = acc11[r];
        if (RELU_BIAS) {
            v00 += bv0; v01 += bv1; v10 += bv0; v11 += bv1;
            v00 = v00 > 0.0f ? v00 : 0.0f;
            v01 = v01 > 0.0f ? v01 : 0.0f;
            v10 = v10 > 0.0f ? v10 : 0.0f;
            v11 = v11 > 0.0f ? v11 : 0.0f;
        }
        C[(size_t)row0 * N + n0 + l16]      = v00;
        C[(size_t)row0 * N + n0 + 16 + l16] = v01;
        C[(size_t)row1 * N + n0 + l16]      = v10;
        C[(size_t)row1 * N + n0 + 16 + l16] = v11;
    }
}

// ---------------------------------------------------------------------------
// Causal mean pooling: xavg[b,i,d] = (sum_{t<=i} q[b,t,d]) / (i+1)
// ---------------------------------------------------------------------------
__global__ __launch_bounds__(256) void cumavg_kernel(
    const float* __restrict__ q, float* __restrict__ xavg, int Bsz)
{
    int idx = blockIdx.x * blockDim.x + threadIdx.x;
    if (idx >= Bsz * D_DIM) return;
    int b = idx / D_DIM, d = idx % D_DIM;
    size_t base = (size_t)b * LSEQ * D_DIM + d;
    float acc = 0.0f;
    for (int i = 0; i < LSEQ; ++i) {
        acc += q[base + (size_t)i * D_DIM];
        xavg[base + (size_t)i * D_DIM] = acc / (float)(i + 1);
    }
}

// ---------------------------------------------------------------------------
// Suffix row sums: S[b*L+i] = sum_{k=L-1-i}^{L-1} g[b,i,k] + 1e-8 (wave/row)
// ---------------------------------------------------------------------------
__global__ __launch_bounds__(256) void rowsum_kernel(
    const float* __restrict__ g, float* __restrict__ S, int nrows)
{
    int wid  = (int)((blockIdx.x * blockDim.x + threadIdx.x) >> 5);
    int lane = threadIdx.x & 31;
    if (wid >= nrows) return;
    int i = wid % LSEQ;
    const float* row = g + (size_t)wid * LSEQ;
    float s = 0.0f;
    for (int k = (LSEQ - 1 - i) + lane; k < LSEQ; k += 32) s += row[k];
#pragma unroll
    for (int off = 16; off; off >>= 1) s += __shfl_xor(s, off, 32);
    if (lane == 0) S[wid] = s + 1e-8f;
}

// ---------------------------------------------------------------------------
// out[b,i,d] = (1/S_i) * sum_{j<=i} g[b,i, L-1-i+j] * q[b,j,d]
// One wave -> 16 rows x 32 cols (A frag reused for 2 B frags). The K-loop is
// split: unmasked bulk (all j < i0 <= every lane's iA) + 4 masked diagonal
// steps. EXEC is all-ones at every WMMA.
// ---------------------------------------------------------------------------
__global__ __launch_bounds__(128) void attn_apply_wmma(
    const float* __restrict__ g, const float* __restrict__ q,
    const float* __restrict__ S, float* __restrict__ out)
{
    const int lane = threadIdx.x & 31;
    const int wave = threadIdx.x >> 5;
    const int b    = blockIdx.z;
    const int i0   = blockIdx.y * 16;
    const int d0   = blockIdx.x * 128 + wave * 32;
    const int half = lane >> 4;
    const int l16  = lane & 15;
    const int koff = half * 2;

    const int iA    = i0 + l16;        // this lane's A row
    const int shift = LSEQ - 1 - iA;   // attn col j -> g col shift+j
    const float* grow = g + ((size_t)b * LSEQ + iA) * LSEQ + shift;
    const float* qB0  = q + (size_t)b * LSEQ * D_DIM + d0 + l16;
    const float* qB1  = qB0 + 16;

    v8f acc0 = {}, acc1 = {};

    // Bulk: j0+3 <= i0-1 < iA for every lane -> no masking needed.
    for (int j0 = 0; j0 < i0; j0 += 4) {
        int jx = j0 + koff;
        v2f a, b0, b1;
        a.x = grow[jx];
        a.y = grow[jx + 1];
        size_t r0 = (size_t)jx * D_DIM;
        size_t r1 = r0 + D_DIM;
        b0.x = qB0[r0];  b0.y = qB0[r1];
        b1.x = qB1[r0];  b1.y = qB1[r1];
        acc0 = __builtin_amdgcn_wmma_f32_16x16x4_f32(false, a, false, b0, (short)0, acc0, false, false);
        acc1 = __builtin_amdgcn_wmma_f32_16x16x4_f32(false, a, false, b1, (short)0, acc1, false, false);
    }

    // Diagonal tail: 4 masked steps covering j in [i0, i0+16).
#pragma unroll
    for (int t = 0; t < 4; ++t) {
        int jx = i0 + t * 4 + koff;
        int jy = jx + 1;
        v2f a, b0, b1;
        a.x = (jx <= iA) ? grow[jx] : 0.0f;
        a.y = (jy <= iA) ? grow[jy] : 0.0f;
        size_t r0 = (size_t)jx * D_DIM;
        size_t r1 = r0 + D_DIM;
        b0.x = qB0[r0];  b0.y = qB0[r1];
        b1.x = qB1[r0];  b1.y = qB1[r1];
        acc0 = __builtin_amdgcn_wmma_f32_16x16x4_f32(false, a, false, b0, (short)0, acc0, false, false);
        acc1 = __builtin_amdgcn_wmma_f32_16x16x4_f32(false, a, false, b1, (short)0, acc1, false, false);
    }

#pragma unroll
    for (int r = 0; r < 8; ++r) {
        int row = i0 + r + half * 8;
        float invs = 1.0f / S[b * LSEQ + row];
        size_t o = ((size_t)b * LSEQ + row) * D_DIM + d0 + l16;
        out[o]      = acc0[r] * invs;
        out[o + 16] = acc1[r] * invs;
    }
}

// ---------------------------------------------------------------------------
extern "C" void kernel_launch(void* const* d_in, const int* in_sizes, int n_in,
                              void* d_out, int out_size, void* d_ws, size_t ws_size,
                              hipStream_t stream)
{
    const float* x  = (const float*)d_in[0];   // [B, L, D]
    const float* Wq = (const float*)d_in[1];   // [D, D]
    const float* Wg = (const float*)d_in[2];   // [D, MAX_LEN]
    const float* bg = (const float*)d_in[3];   // [MAX_LEN]
    const float* Wc = (const float*)d_in[4];   // [D, D]
    float* out = (float*)d_out;

    const int Bsz = in_sizes[0] / (LSEQ * D_DIM);
    const int M   = Bsz * LSEQ;                // 8192 flattened rows

    // Workspace carve-out (floats): q | xavg(->tmp) | g | S   (~128 MB)
    float* q    = (float*)d_ws;
    float* xavg = q    + (size_t)M * D_DIM;
    float* g    = xavg + (size_t)M * D_DIM;
    float* S    = g    + (size_t)M * LSEQ;

    dim3 blk(128);

    // 1) q = x @ Wq
    gemm_f32_wmma<false><<<dim3(D_DIM / 64, M / 64), blk, 0, stream>>>(
        x, Wq, nullptr, q, M, D_DIM, D_DIM);

    // 2) causal mean pooling
    cumavg_kernel<<<(Bsz * D_DIM + 255) / 256, 256, 0, stream>>>(q, xavg, Bsz);

    // 3) g = relu(xavg @ Wg + bg)
    gemm_f32_wmma<true><<<dim3(LSEQ / 64, M / 64), blk, 0, stream>>>(
        xavg, Wg, bg, g, M, LSEQ, D_DIM);

    // 4) shifted-row sums for normalization
    rowsum_kernel<<<(M * 32 + 255) / 256, 256, 0, stream>>>(g, S, M);

    // 5) tmp = attn @ q   (tmp overwrites xavg, which is dead now)
    attn_apply_wmma<<<dim3(D_DIM / 128, LSEQ / 16, Bsz), blk, 0, stream>>>(
        g, q, S, xavg);

    // 6) out = tmp @ Wc
    gemm_f32_wmma<false><<<dim3(D_DIM / 64, M / 64), blk, 0, stream>>>(
        xavg, Wc, nullptr, out, M, D_DIM, D_DIM);
}